// ResidualBlock_18743237280519
// MI455X (gfx1250) — compile-verified
//
#include <hip/hip_runtime.h>
#include <hip/hip_bf16.h>

typedef float v2f __attribute__((ext_vector_type(2)));
typedef float v8f __attribute__((ext_vector_type(8)));

#define D 128
#define SLOPE 0.1f
#define BN_EPS 1e-5f
#define XS_STRIDE 132   // 128 + 4 pad -> conflict-free 8B LDS loads

__device__ __forceinline__ float leaky(float v) {
    // slope < 1: leaky_relu(v) == max(v, slope*v)
    return fmaxf(v, SLOPE * v);
}

// ---------- degree / normalization ----------
__global__ void k_init_deg(float* deg, int N) {
    int i = blockIdx.x * blockDim.x + threadIdx.x;
    if (i < N) deg[i] = 1.0f;            // self-loop contributes 1
}

__global__ void k_count_deg(const int* __restrict__ row, float* deg, int E) {
    int e = blockIdx.x * blockDim.x + threadIdx.x;
    if (e < E) atomicAdd(&deg[row[e]], 1.0f);
}

// dis = rsqrt(deg) in place; deg2 initialized with self-loop terms (dis^2 + 2)
__global__ void k_dis_deg2init(float* deg_dis, float* deg2, int N) {
    int i = blockIdx.x * blockDim.x + threadIdx.x;
    if (i < N) {
        float dis = rsqrtf(deg_dis[i]);  // deg >= 1 always
        deg_dis[i] = dis;
        deg2[i] = dis * dis + 2.0f;
    }
}

__global__ void k_deg2_scatter(const int* __restrict__ row, const int* __restrict__ col,
                               const float* __restrict__ dis, float* deg2, int E) {
    int e = blockIdx.x * blockDim.x + threadIdx.x;
    if (e < E) {
        int r = row[e], c = col[e];
        atomicAdd(&deg2[c], dis[r] * dis[c]);
    }
}

__global__ void k_dis2(float* deg2_dis2, int N) {
    int i = blockIdx.x * blockDim.x + threadIdx.x;
    if (i < N) deg2_dis2[i] = rsqrtf(deg2_dis2[i]);   // deg2 >= 2 always
}

// ---------- h = x @ W^T via V_WMMA_F32_16X16X4_F32 ----------
// Block: 256 threads = 8 waves; block computes a 16-row x 128-col strip of h.
// Wave w owns the 16-column tile [16w, 16w+16). K-loop: 32 chained WMMAs.
__global__ __launch_bounds__(256) void k_gemm_wmma(const float* __restrict__ x,
                                                   const float* __restrict__ W,
                                                   float* __restrict__ h, int N) {
    __shared__ float xs[16 * XS_STRIDE];
    const int row0 = blockIdx.x * 16;

    // Stage the 16x128 x-strip into LDS (padded stride 132).
    for (int i = threadIdx.x; i < 16 * D; i += 256) {
        int r = i >> 7, c = i & (D - 1);
        int rr = row0 + r;
        if (rr >= N) rr = N - 1;
        xs[r * XS_STRIDE + c] = x[(size_t)rr * D + c];
    }
    __syncthreads();

    const int wave = threadIdx.x >> 5;     // 0..7 -> column tile
    const int lane = threadIdx.x & 31;
    const int hi   = lane >> 4;            // lane half selects K sub-pair
    const int lm   = lane & 15;
    const int ncol = wave * 16 + lm;       // output column / W row

    const float* __restrict__ wrow = W + (size_t)ncol * D;   // W[n][k], row-major

    v8f acc = {};
    #pragma unroll
    for (int k = 0; k < D; k += 4) {
        const int kk = k + 2 * hi;
        // A (16x4 f32): lane half 0 -> K = k,k+1 ; half 1 -> K = k+2,k+3 ; M = lm
        v2f a = { xs[lm * XS_STRIDE + kk], xs[lm * XS_STRIDE + kk + 1] };
        // B (4x16 f32): mirrored layout, N = lm per half
        v2f b = { wrow[kk], wrow[kk + 1] };
        acc = __builtin_amdgcn_wmma_f32_16x16x4_f32(
            /*neg_a=*/false, a, /*neg_b=*/false, b,
            /*c_mod=*/(short)0, acc, /*reuse_a=*/false, /*reuse_b=*/false);
    }

    // C/D layout: VGPR v -> M = v + 8*hi, N = lm
    #pragma unroll
    for (int v = 0; v < 8; ++v) {
        int r = row0 + v + 8 * hi;
        if (r < N) h[(size_t)r * D + ncol] = acc[v];
    }
}

// ---------- aggregation ----------
// agg[i,:] = bias + (dis[i]^2 + 2) * dis2[i]^2 * h[i,:]   (both self-loop passes folded in)
// Vectorized: one float4 per thread -> b128 load/store, streaming-limited.
__global__ void k_init_agg(const float4* __restrict__ h4, const float4* __restrict__ bias4,
                           const float* __restrict__ dis, const float* __restrict__ dis2,
                           float4* __restrict__ agg4, int N) {
    size_t idx = (size_t)blockIdx.x * blockDim.x + threadIdx.x;   // over N*32 float4s
    if (idx >= (size_t)N * (D / 4)) return;
    int n = (int)(idx >> 5);
    int q = (int)(idx & 31);
    float di = dis[n], d2 = dis2[n];
    float s = (di * di + 2.0f) * d2 * d2;
    float4 hv = h4[idx];
    float4 bv = bias4[q];
    float4 r;
    r.x = fmaf(s, hv.x, bv.x);
    r.y = fmaf(s, hv.y, bv.y);
    r.z = fmaf(s, hv.z, bv.z);
    r.w = fmaf(s, hv.w, bv.w);
    agg4[idx] = r;
}

// One wave per edge: 8 edges per 256-thread block. Each lane gathers one float4
// of h[row] (global_load_b128, 512B coalesced per wave) and issues 4 f32 atomics
// into agg[col]. Norm scalars loaded once per wave (uniform broadcast).
__global__ __launch_bounds__(256) void k_edge_scatter(const int* __restrict__ row,
                                                      const int* __restrict__ col,
                                                      const float* __restrict__ dis,
                                                      const float* __restrict__ dis2,
                                                      const float* __restrict__ h,
                                                      float* __restrict__ agg, int E) {
    int e = blockIdx.x * 8 + (threadIdx.x >> 5);
    int lane = threadIdx.x & 31;
    if (e >= E) return;
    int r = row[e], c = col[e];
    float s = dis2[r] * dis[r] * dis[c] * dis2[c];
    float4 v = ((const float4*)(h + (size_t)r * D))[lane];
    float* ac = agg + (size_t)c * D + lane * 4;
    atomicAdd(ac + 0, s * v.x);
    atomicAdd(ac + 1, s * v.y);
    atomicAdd(ac + 2, s * v.z);
    atomicAdd(ac + 3, s * v.w);
}

// ---------- batch norm ----------
__global__ void k_zero_stats(float* sums, float* sumsq) {
    int d = threadIdx.x;
    if (d < 2 * D) sums[d] = 0.0f;       // sums and sumsq are contiguous
}

// 32 threads/row-group x float4 columns: each thread owns 4 adjacent columns.
__global__ __launch_bounds__(128) void k_bn_reduce(const float4* __restrict__ agg4,
                                                   float* sums, float* sumsq, int N) {
    int q = threadIdx.x & 31;            // float4 column group 0..31
    int sub = threadIdx.x >> 5;          // 4 row-phases per block
    float4 s = {0.f, 0.f, 0.f, 0.f}, sq = {0.f, 0.f, 0.f, 0.f};
    for (int n = blockIdx.x * 4 + sub; n < N; n += gridDim.x * 4) {
        float4 v = agg4[(size_t)n * (D / 4) + q];
        s.x += v.x; s.y += v.y; s.z += v.z; s.w += v.w;
        sq.x += v.x * v.x; sq.y += v.y * v.y; sq.z += v.z * v.z; sq.w += v.w * v.w;
    }
    int d = q * 4;
    atomicAdd(&sums[d + 0], s.x);  atomicAdd(&sums[d + 1], s.y);
    atomicAdd(&sums[d + 2], s.z);  atomicAdd(&sums[d + 3], s.w);
    atomicAdd(&sumsq[d + 0], sq.x); atomicAdd(&sumsq[d + 1], sq.y);
    atomicAdd(&sumsq[d + 2], sq.z); atomicAdd(&sumsq[d + 3], sq.w);
}

__global__ void k_bn_params(const float* __restrict__ sums, const float* __restrict__ sumsq,
                            const float* __restrict__ gamma, const float* __restrict__ beta,
                            float* scale, float* shift, int N) {
    int d = threadIdx.x;
    if (d >= D) return;
    float invN = 1.0f / (float)N;
    float mu = sums[d] * invN;
    float var = sumsq[d] * invN - mu * mu;
    float inv = rsqrtf(var + BN_EPS);
    float sc = gamma[d] * inv;
    scale[d] = sc;
    shift[d] = beta[d] - mu * sc;
}

// out = leaky( leaky(agg*scale + shift) + x ), float4-vectorized streaming pass
__global__ void k_finalize(const float4* __restrict__ agg4, const float4* __restrict__ x4,
                           const float4* __restrict__ scale4, const float4* __restrict__ shift4,
                           float4* __restrict__ out4, int N) {
    size_t idx = (size_t)blockIdx.x * blockDim.x + threadIdx.x;   // over N*32 float4s
    if (idx >= (size_t)N * (D / 4)) return;
    int q = (int)(idx & 31);
    float4 a = agg4[idx];
    float4 sc = scale4[q];
    float4 sh = shift4[q];
    float4 xr = x4[idx];
    float4 o;
    o.x = leaky(leaky(fmaf(a.x, sc.x, sh.x)) + xr.x);
    o.y = leaky(leaky(fmaf(a.y, sc.y, sh.y)) + xr.y);
    o.z = leaky(leaky(fmaf(a.z, sc.z, sh.z)) + xr.z);
    o.w = leaky(leaky(fmaf(a.w, sc.w, sh.w)) + xr.w);
    out4[idx] = o;
}

// ---------- host launcher ----------
extern "C" void kernel_launch(void* const* d_in, const int* in_sizes, int n_in,
                              void* d_out, int out_size, void* d_ws, size_t ws_size,
                              hipStream_t stream) {
    const float* x     = (const float*)d_in[0];
    const float* W     = (const float*)d_in[1];
    const float* bias  = (const float*)d_in[2];
    const float* gamma = (const float*)d_in[3];
    const float* beta  = (const float*)d_in[4];
    const int*   ei    = (const int*)d_in[5];

    const int N = in_sizes[0] / D;
    const int E = in_sizes[5] / 2;
    const int* row = ei;          // edge_index[0,:]
    const int* col = ei + E;      // edge_index[1,:]

    float* dis   = (float*)d_ws;                 // N  (deg -> dis in place)
    float* dis2  = dis + N;                      // N  (deg2 -> dis2 in place)
    float* h     = dis2 + N;                     // N*D
    float* agg   = h + (size_t)N * D;            // N*D
    float* sums  = agg + (size_t)N * D;          // D
    float* sumsq = sums + D;                     // D (contiguous with sums)
    float* scale = sumsq + D;                    // D
    float* shift = scale + D;                    // D
    float* out   = (float*)d_out;

    const int T = 256;
    const int nodeBlocks = (N + T - 1) / T;
    const int edgeBlocks = (E + T - 1) / T;
    const size_t vec4 = (size_t)N * (D / 4);
    const int vecBlocks = (int)((vec4 + T - 1) / T);

    // degree / norm passes
    k_init_deg<<<nodeBlocks, T, 0, stream>>>(dis, N);
    k_count_deg<<<edgeBlocks, T, 0, stream>>>(row, dis, E);
    k_dis_deg2init<<<nodeBlocks, T, 0, stream>>>(dis, dis2, N);
    k_deg2_scatter<<<edgeBlocks, T, 0, stream>>>(row, col, dis, dis2, E);
    k_dis2<<<nodeBlocks, T, 0, stream>>>(dis2, N);

    // GEMM: h = x @ W^T
    const int gemmBlocks = (N + 15) / 16;
    k_gemm_wmma<<<gemmBlocks, 256, 0, stream>>>(x, W, h, N);

    // aggregation
    k_init_agg<<<vecBlocks, T, 0, stream>>>((const float4*)h, (const float4*)bias,
                                            dis, dis2, (float4*)agg, N);
    k_edge_scatter<<<(E + 7) / 8, 256, 0, stream>>>(row, col, dis, dis2, h, agg, E);

    // batch norm + activation + residual
    k_zero_stats<<<1, 256, 0, stream>>>(sums, sumsq);
    k_bn_reduce<<<512, 128, 0, stream>>>((const float4*)agg, sums, sumsq, N);
    k_bn_params<<<1, 128, 0, stream>>>(sums, sumsq, gamma, beta, scale, shift, N);
    k_finalize<<<vecBlocks, T, 0, stream>>>((const float4*)agg, (const float4*)x,
                                            (const float4*)scale, (const float4*)shift,
                                            (float4*)out, N);
}